// Plugboard_25443386261689
// MI455X (gfx1250) — compile-verified
//
#include <hip/hip_runtime.h>
#include <hip/hip_bf16.h>
#include <cstdint>

// Problem constants (from reference): y = x @ (W ⊙ I)^T  ==  x * diag(W)
constexpr int Dv = 2048;
constexpr int Bv = 16384;

constexpr int TPB            = 256;  // 8 wave32 per block
constexpr int ROWS_PER_BLOCK = 16;   // rows of x handled per block
// each block covers TPB float4 columns = 1024 scalar columns -> gridDim.x = 2

#define AS_GLOBAL __attribute__((address_space(1)))
#define AS_LDS    __attribute__((address_space(3)))

// Pointee type the async builtins expect (per hipcc diagnostic):
// 'int __attribute__((vector_size(16)))'
typedef int v4i __attribute__((vector_size(16)));

// ---- CDNA5 async global->LDS copy (ASYNCcnt path) ----------------------
__device__ __forceinline__ void async_g2lds_b128(const void* gptr, void* lptr) {
#if __has_builtin(__builtin_amdgcn_global_load_async_to_lds_b128)
    __builtin_amdgcn_global_load_async_to_lds_b128(
        (AS_GLOBAL v4i*)(uintptr_t)gptr,
        (AS_LDS v4i*)(uintptr_t)lptr,
        /*offset=*/0, /*cpol=*/0);
#else
    // VDST = LDS byte address (low 32 bits of generic LDS pointer),
    // VADDR = 64-bit global address, saddr = off (GV mode).
    unsigned lds_off = (unsigned)(uintptr_t)lptr;
    asm volatile("global_load_async_to_lds_b128 %0, %1, off"
                 :
                 : "v"(lds_off), "v"(gptr)
                 : "memory");
#endif
}

__device__ __forceinline__ void wait_async_zero() {
#if __has_builtin(__builtin_amdgcn_s_wait_asynccnt)
    __builtin_amdgcn_s_wait_asynccnt(0);
#else
    asm volatile("s_wait_asynccnt 0" ::: "memory");
#endif
}

// ---- Kernel: out[b, d] = x[b, d] * scale[d]  (HBM-streaming) -----------
// Defined FIRST so the disasm snippet shows the async-tensor sequence.
__global__ __launch_bounds__(TPB) void scale_rows_kernel(
    const float4* __restrict__ x, const float4* __restrict__ scale4,
    float4* __restrict__ out) {
    __shared__ float4 s_scale[TPB];  // this block's 4 KB slice of diag scale

    const int c4      = blockIdx.x * TPB + threadIdx.x;  // float4 column, 0..511
    const int rowBase = blockIdx.y * ROWS_PER_BLOCK;

    // Stage the scale slice into LDS via the CDNA5 async path. Each lane
    // copies (and later reads back) its own 16 bytes, so ASYNCcnt==0 is the
    // only hard requirement; the barrier also gives the compiler ordering.
    async_g2lds_b128(&scale4[c4], &s_scale[threadIdx.x]);
    wait_async_zero();
    __syncthreads();

    const float4 s = s_scale[threadIdx.x];  // ds_load_b128

    constexpr int STRIDE4 = Dv / 4;  // 512 float4 per row
    const size_t base = (size_t)rowBase * STRIDE4 + (size_t)c4;

    // Fully unrolled: 16 independent b128 loads in flight per wave, then
    // 16 b128 stores. 128 KB moved per block; pure bandwidth streaming.
    float4 v[ROWS_PER_BLOCK];
#pragma unroll
    for (int r = 0; r < ROWS_PER_BLOCK; ++r) {
        v[r] = x[base + (size_t)r * STRIDE4];
    }
#pragma unroll
    for (int r = 0; r < ROWS_PER_BLOCK; ++r) {
        float4 o;
        o.x = v[r].x * s.x;
        o.y = v[r].y * s.y;
        o.z = v[r].z * s.z;
        o.w = v[r].w * s.w;
        out[base + (size_t)r * STRIDE4] = o;
    }
}

// ---- Kernel: scale[d] = W[d,d] * M[d,d]  (8 KB result in d_ws) ---------
__global__ __launch_bounds__(TPB) void diag_scale_kernel(
    const float* __restrict__ W, const float* __restrict__ M,
    float* __restrict__ scale) {
    int d = blockIdx.x * blockDim.x + threadIdx.x;
    if (d < Dv) {
        size_t idx = (size_t)d * (size_t)(Dv + 1);  // d*D + d
        scale[d] = W[idx] * M[idx];
    }
}

extern "C" void kernel_launch(void* const* d_in, const int* in_sizes, int n_in,
                              void* d_out, int out_size, void* d_ws, size_t ws_size,
                              hipStream_t stream) {
    const float* x = (const float*)d_in[0];     // [B, D]
    const float* W = (const float*)d_in[1];     // [D, D]
    const float* M = (const float*)d_in[2];     // [D, D] (identity)
    float* out     = (float*)d_out;             // [B, D]
    float* scale   = (float*)d_ws;              // D floats (8 KB) scratch

    diag_scale_kernel<<<Dv / TPB, TPB, 0, stream>>>(W, M, scale);

    dim3 grid(Dv / (4 * TPB), Bv / ROWS_PER_BLOCK);  // (2, 1024)
    scale_rows_kernel<<<grid, TPB, 0, stream>>>(
        (const float4*)x, (const float4*)scale, (float4*)out);
}